// MCMCLoss_71159018160363
// MI455X (gfx1250) — compile-verified
//
#include <hip/hip_runtime.h>
#include <hip/hip_bf16.h>
#include <math.h>

#define BN   2048
#define DD   1024
#define NN   4096   // 2*BN

typedef __attribute__((ext_vector_type(2))) float v2f;
typedef __attribute__((ext_vector_type(8))) float v8f;
typedef __attribute__((ext_vector_type(4))) unsigned int v4u;
typedef __attribute__((ext_vector_type(8))) int v8i;
typedef __attribute__((ext_vector_type(4))) int v4i;

// ---------------------------------------------------------------------------
// helpers
// ---------------------------------------------------------------------------
__device__ __forceinline__ const float* hrow(const float* hi, const float* hj, int r) {
    return (r < BN) ? (hi + (size_t)r * DD) : (hj + (size_t)(r - BN) * DD);
}

__device__ __forceinline__ bool better(float v1, int i1, float v2, int i2) {
    return (v1 > v2) || (v1 == v2 && i1 < i2);   // jax top_k tie-break: lower index
}

// ---------------------------------------------------------------------------
// TDM: DMA a 128-row x 16-col fp32 tile (row stride DD) global -> LDS.
// D# per CDNA5 ISA: pad_enable adds 1 DWORD after every 16 DWORDs, i.e. the
// tile lands in LDS with row pitch 17 floats = conflict-free fragment reads.
// clang-23 toolchain: 6-arg form (g0 v4u, g1 v8i, v4i, v4i, v8i, i32 cpol).
// ---------------------------------------------------------------------------
__device__ __forceinline__ void tdm_load_tile128x16(const float* gsrc, unsigned lds_off) {
    unsigned long long ga = (unsigned long long)(size_t)gsrc;
    v4u g0;
    g0[0] = 1u;                                            // count=1 (valid D#)
    g0[1] = lds_off;                                       // lds_addr (bytes)
    g0[2] = (unsigned)(ga & 0xffffffffu);                  // global_addr[31:0]
    g0[3] = (unsigned)((ga >> 32) & 0x01ffffffu)           // global_addr[56:32]
          | (2u << 30);                                    // type=2 ("image")
    v8i g1;
    g1[0] = (int)(0x00020000u                              // data_size=2 -> 4B
          | (1u << 20)                                     // pad_enable
          | (3u << 22));                                   // pad_interval: 16 DWORDs
                                                           // pad_amount=0 -> 1 DWORD
    g1[1] = (int)(1024u << 16);                            // tensor_dim0 = DD
    g1[2] = (int)(2048u << 16);                            // tensor_dim1 = BN rows
    g1[3] = (int)(16u << 16);                              // tile_dim0 = 16
    g1[4] = 128;                                           // tile_dim1 = 128 rows
    g1[5] = 1024;                                          // tensor_dim0_stride = DD
    g1[6] = 0;
    g1[7] = 0;
    v4i z4 = {0, 0, 0, 0};                                 // groups 2/3 unused (2D)
    v8i z8 = {0, 0, 0, 0, 0, 0, 0, 0};
    __builtin_amdgcn_tensor_load_to_lds(g0, g1, z4, z4, z8, 0);
}

// ---------------------------------------------------------------------------
// Kernel 1: sim = (h h^T) * 2 via fp32 WMMA + TDM double-buffered staging.
// 128x128 tile per block, 256 thr = 8 waves, wave -> 32x64 patch (2x4 frags)
// ---------------------------------------------------------------------------
__global__ __launch_bounds__(256) void sim_gemm_f32wmma(
    const float* __restrict__ hi, const float* __restrict__ hj,
    float* __restrict__ sim)
{
    __shared__ float As[2][128][17];   // TDM pad -> pitch 17
    __shared__ float Bs[2][128][17];

    const int t    = threadIdx.x;
    const int wave = t >> 5;
    const int lane = t & 31;
    const int hl   = lane >> 4;        // half-wave select
    const int lm   = lane & 15;
    const int wm   = (wave & 3) * 32;  // wave M offset in tile
    const int wn   = (wave >> 2) * 64; // wave N offset in tile
    const int bM   = blockIdx.y * 128;
    const int bN   = blockIdx.x * 128;

    // 128-row tiles never straddle the hi/hj boundary (BN % 128 == 0)
    const float* aBase = hrow(hi, hj, bM);
    const float* bBase = hrow(hi, hj, bN);
    const unsigned ldsA[2] = { (unsigned)(size_t)&As[0][0][0],
                               (unsigned)(size_t)&As[1][0][0] };
    const unsigned ldsB[2] = { (unsigned)(size_t)&Bs[0][0][0],
                               (unsigned)(size_t)&Bs[1][0][0] };

    v8f acc[2][4] = {};

    // prologue: stage 0 into buffer 0
    if (t == 0) {
        tdm_load_tile128x16(aBase, ldsA[0]);
        tdm_load_tile128x16(bBase, ldsB[0]);
        __builtin_amdgcn_s_wait_tensorcnt(0);
    }
    __syncthreads();

    int buf = 0;
    for (int kt = 0; kt < DD; kt += 16) {
        // prefetch next stage into the other buffer (async, TENSORcnt)
        if (t == 0 && kt + 16 < DD) {
            tdm_load_tile128x16(aBase + kt + 16, ldsA[buf ^ 1]);
            tdm_load_tile128x16(bBase + kt + 16, ldsB[buf ^ 1]);
        }

        #pragma unroll
        for (int kk = 0; kk < 16; kk += 4) {
            v2f a[2], b[4];
            #pragma unroll
            for (int mi = 0; mi < 2; ++mi) {
                const float* p = &As[buf][wm + mi*16 + lm][kk + 2*hl];
                a[mi].x = p[0]; a[mi].y = p[1];   // lanes0-15: K=k,k+1 ; 16-31: K=k+2,k+3
            }
            #pragma unroll
            for (int ni = 0; ni < 4; ++ni) {
                const float* p = &Bs[buf][wn + ni*16 + lm][kk + 2*hl];
                b[ni].x = p[0]; b[ni].y = p[1];
            }
            #pragma unroll
            for (int mi = 0; mi < 2; ++mi)
                #pragma unroll
                for (int ni = 0; ni < 4; ++ni)
                    acc[mi][ni] = __builtin_amdgcn_wmma_f32_16x16x4_f32(
                        false, a[mi], false, b[ni], (short)0, acc[mi][ni], false, false);
        }

        if (t == 0) __builtin_amdgcn_s_wait_tensorcnt(0);  // next stage landed
        __syncthreads();
        buf ^= 1;
    }

    // C layout: VGPR v -> (M = v + 8*hl, N = lm); scale by 1/temp = 2
    #pragma unroll
    for (int mi = 0; mi < 2; ++mi)
        #pragma unroll
        for (int ni = 0; ni < 4; ++ni) {
            const int rg = bM + wm + mi*16 + 8*hl;
            const int cg = bN + wn + ni*16 + lm;
            #pragma unroll
            for (int v = 0; v < 8; ++v)
                sim[(size_t)(rg + v) * NN + cg] = acc[mi][ni][v] * 2.0f;
        }
}

// ---------------------------------------------------------------------------
// Kernel 2: per-anchor exact top-4 of concate row + acceptance
// concate[j] = j<B ? sim[i][B+j] : sim[B+i][j-B]
// ---------------------------------------------------------------------------
__global__ __launch_bounds__(256) void topk_accept(
    const float* __restrict__ sim, int* __restrict__ e, int* __restrict__ xc)
{
    __shared__ float lv[256][4];
    __shared__ int   li[256][4];
    const int i = blockIdx.x;
    const int t = threadIdx.x;

    float tv[4]; int ti[4];
    for (int k = 0; k < 4; ++k) { tv[k] = -INFINITY; ti[k] = 0x7FFFFFFF; }

    for (int j = t; j < NN; j += 256) {
        float v = (j < BN) ? sim[(size_t)i * NN + BN + j]
                           : sim[(size_t)(BN + i) * NN + (j - BN)];
        if (better(v, j, tv[3], ti[3])) {
            tv[3] = v; ti[3] = j;
            for (int k = 3; k > 0; --k) {
                if (better(tv[k], ti[k], tv[k-1], ti[k-1])) {
                    float fv = tv[k]; tv[k] = tv[k-1]; tv[k-1] = fv;
                    int  ii = ti[k]; ti[k] = ti[k-1]; ti[k-1] = ii;
                } else break;
            }
        }
    }
    for (int k = 0; k < 4; ++k) { lv[t][k] = tv[k]; li[t][k] = ti[k]; }
    __syncthreads();

    for (int s = 128; s > 0; s >>= 1) {
        if (t < s) {
            float av[4], bv[4], ov[4]; int ai[4], bi[4], oi[4];
            for (int k = 0; k < 4; ++k) {
                av[k] = lv[t][k];   ai[k] = li[t][k];
                bv[k] = lv[t+s][k]; bi[k] = li[t+s][k];
            }
            int ia = 0, ib = 0;
            for (int o = 0; o < 4; ++o) {
                if (better(av[ia], ai[ia], bv[ib], bi[ib])) { ov[o]=av[ia]; oi[o]=ai[ia]; ++ia; }
                else                                        { ov[o]=bv[ib]; oi[o]=bi[ib]; ++ib; }
            }
            for (int k = 0; k < 4; ++k) { lv[t][k] = ov[k]; li[t][k] = oi[k]; }
        }
        __syncthreads();
    }

    if (t == 0) {
        int ci[2]  = {0x7FFFFFFF, 0x7FFFFFFF};   // extras of row i   (cols >= B)
        int cbi[2] = {0x7FFFFFFF, 0x7FFFFFFF};   // extras of row B+i (cols <  B)
        int ni = 0, nbi = 0, acc = 0;
        for (int k = 0; k < 4 && acc < 2; ++k) {
            int idx = li[0][k];
            if (idx == i || idx == i + BN) continue;   // invalid (anchor pair)
            ++acc;
            if (idx >= BN) cbi[nbi++] = idx - BN;      // row B+i, col idx-B
            else           ci[ni++]   = BN + idx;      // row i,   col B+idx
        }
        e[i]      = ni;
        e[BN + i] = nbi;
        xc[2*i]          = ci[0];  xc[2*i + 1]          = ci[1];
        xc[2*(BN+i)]     = cbi[0]; xc[2*(BN+i) + 1]     = cbi[1];
    }
}

// ---------------------------------------------------------------------------
// Kernel 3: exclusive prefix E of e; build globally-sorted positive flat list
// ---------------------------------------------------------------------------
__global__ __launch_bounds__(256) void scan_pos(
    const int* __restrict__ e, const int* __restrict__ xc,
    int* __restrict__ E, int* __restrict__ pf)
{
    __shared__ int part[257];
    const int t = threadIdx.x;
    int s = 0;
    for (int r = t*16; r < t*16 + 16; ++r) s += e[r];
    part[t + 1] = s;
    if (t == 0) part[0] = 0;
    __syncthreads();
    if (t == 0) for (int k = 1; k <= 256; ++k) part[k] += part[k-1];
    __syncthreads();

    int run = part[t];
    for (int r = t*16; r < t*16 + 16; ++r) {
        E[r] = run;
        int pair = (r < BN) ? r + BN : r - BN;
        int x0 = pair, x1 = xc[2*r], x2 = xc[2*r + 1];   // absent = INT_MAX
        int tmp;
        if (x0 > x1) { tmp=x0; x0=x1; x1=tmp; }
        if (x1 > x2) { tmp=x1; x1=x2; x2=tmp; }
        if (x0 > x1) { tmp=x0; x0=x1; x1=tmp; }
        const int C = r + run;            // global pos rank offset of row r
        const int cnt = 1 + e[r];
        int xs[3] = {x0, x1, x2};
        for (int k = 0; k < cnt; ++k) pf[C + k] = r * NN + xs[k];
        run += e[r];
    }
}

// ---------------------------------------------------------------------------
// Kernel 4: per output row r, logsumexp over { posFlat[2r],posFlat[2r+1] } U
// global neg ranks [(N-3)r,(N-3)(r+1)) mapped via rowNegStart(s)=s(N-2)-E[s]
// ---------------------------------------------------------------------------
__device__ __forceinline__ long long rns(int s, const int* E) {
    return (long long)s * (NN - 2) - E[s];
}

__global__ __launch_bounds__(256) void group_loss_k(
    const float* __restrict__ sim, const int* __restrict__ E,
    const int* __restrict__ xc, const int* __restrict__ pf,
    float* __restrict__ gout)
{
    __shared__ long long rstart[8];
    __shared__ int  xs[7][4];
    __shared__ int  srow0;
    __shared__ float pos0v;
    __shared__ float ms[256], ss[256];

    const int r = blockIdx.x;
    const int t = threadIdx.x;
    const long long NEG = NN - 3;
    const long long gs = (long long)r * NEG, ge = gs + NEG;

    if (t == 0) {
        int lo = 0, hi = NN - 1;                      // largest s: rns(s) <= gs
        while (lo < hi) {
            int mid = (lo + hi + 1) >> 1;
            if (rns(mid, E) <= gs) lo = mid; else hi = mid - 1;
        }
        srow0 = lo;
        const long long total = (long long)NN * (NN - 3);
        for (int k = 0; k < 8; ++k) {
            int s = lo + k;
            rstart[k] = (s < NN) ? rns(s, E) : (total + (s - NN));
        }
        for (int k = 0; k < 7; ++k) {
            int s = lo + k;
            int a0 = 0x7FFFFFFF, a1 = a0, a2 = a0, a3 = a0;
            if (s < NN) {
                a0 = s;
                a1 = (s < BN) ? s + BN : s - BN;
                a2 = xc[2*s]; a3 = xc[2*s + 1];
            }
            int tmp;
            if (a0 > a1) { tmp=a0; a0=a1; a1=tmp; }
            if (a2 > a3) { tmp=a2; a2=a3; a3=tmp; }
            if (a0 > a2) { tmp=a0; a0=a2; a2=tmp; }
            if (a1 > a3) { tmp=a1; a1=a3; a3=tmp; }
            if (a1 > a2) { tmp=a1; a1=a2; a2=tmp; }
            xs[k][0]=a0; xs[k][1]=a1; xs[k][2]=a2; xs[k][3]=a3;
        }
        pos0v = sim[pf[2*r]];
    }
    __syncthreads();

    float m = -INFINITY, sum = 0.0f;
    if (t < 2) {                                    // fold in the 2 positives
        m = sim[pf[2*r + t]];
        sum = 1.0f;
    }
    for (long long g = gs + t; g < ge; g += 256) {
        int k = 0;
        while (k < 6 && g >= rstart[k+1]) ++k;
        const int s  = srow0 + k;
        const int tr = (int)(g - rstart[k]);        // rank within row's negs
        int col = tr;                               // skip sorted excluded cols
        if (xs[k][0] <= col) ++col;
        if (xs[k][1] <= col) ++col;
        if (xs[k][2] <= col) ++col;
        if (xs[k][3] <= col) ++col;
        float v = sim[(size_t)s * NN + col];
        if (v > m) { sum = sum * expf(m - v) + 1.0f; m = v; }
        else       { sum += expf(v - m); }
    }
    ms[t] = m; ss[t] = sum;
    __syncthreads();
    for (int s2 = 128; s2 > 0; s2 >>= 1) {
        if (t < s2) {
            float m2 = ms[t + s2], v2 = ss[t + s2];
            float M = fmaxf(ms[t], m2);
            ss[t] = ss[t] * expf(ms[t] - M) + v2 * expf(m2 - M);
            ms[t] = M;
        }
        __syncthreads();
    }
    if (t == 0) gout[r] = ms[0] + logf(ss[0]) - pos0v;
}

// ---------------------------------------------------------------------------
// Kernel 5: deterministic fixed-order reduce, /N
// ---------------------------------------------------------------------------
__global__ __launch_bounds__(256) void final_reduce(
    const float* __restrict__ gout, float* __restrict__ out)
{
    __shared__ float sh[256];
    const int t = threadIdx.x;
    float s = 0.0f;
    for (int i = t; i < NN; i += 256) s += gout[i];
    sh[t] = s;
    __syncthreads();
    for (int k = 128; k > 0; k >>= 1) {
        if (t < k) sh[t] += sh[t + k];
        __syncthreads();
    }
    if (t == 0) out[0] = sh[0] / (float)NN;
}

// ---------------------------------------------------------------------------
extern "C" void kernel_launch(void* const* d_in, const int* in_sizes, int n_in,
                              void* d_out, int out_size, void* d_ws, size_t ws_size,
                              hipStream_t stream) {
    (void)in_sizes; (void)n_in; (void)out_size; (void)ws_size;
    const float* hi = (const float*)d_in[0];
    const float* hj = (const float*)d_in[1];
    float* out = (float*)d_out;

    char* ws = (char*)d_ws;
    float* sim = (float*)ws;
    size_t off = (size_t)NN * NN * sizeof(float);          // 64 MB
    int* e  = (int*)(ws + off); off += NN * sizeof(int);
    int* E  = (int*)(ws + off); off += NN * sizeof(int);
    int* xc = (int*)(ws + off); off += 2 * NN * sizeof(int);
    int* pf = (int*)(ws + off); off += 2 * NN * sizeof(int);
    float* gout = (float*)(ws + off);

    sim_gemm_f32wmma<<<dim3(NN/128, NN/128), 256, 0, stream>>>(hi, hj, sim);
    topk_accept<<<BN, 256, 0, stream>>>(sim, e, xc);
    scan_pos<<<1, 256, 0, stream>>>(e, xc, E, pf);
    group_loss_k<<<NN, 256, 0, stream>>>(sim, E, xc, pf, gout);
    final_reduce<<<1, 256, 0, stream>>>(gout, out);
}